// FeaProp_62096637166372
// MI455X (gfx1250) — compile-verified
//
#include <hip/hip_runtime.h>
#include <math.h>

// ---------------- problem constants (from reference setup_inputs) ------------
#define B_   4
#define N_   4096
#define M_   1024
#define C_   128
#define D_   256
#define PH_  64
#define AH_  512
#define K_   16
#define EPS_ 1e-5f

// padded LDS row strides (words): bank = (4*lane + c) % 64 -> conflict-free b64
#define XS_  (D_ + 4)    // 260 for 256-channel arrays
#define HS_  (PH_ + 4)   // 68  for 64-channel arrays
#define GS_  12          // for the 4-channel h matrix

typedef float v2f __attribute__((ext_vector_type(2)));
typedef float v8f __attribute__((ext_vector_type(8)));

__device__ __forceinline__ v8f zero8() {
  v8f z = {0.f, 0.f, 0.f, 0.f, 0.f, 0.f, 0.f, 0.f};
  return z;
}

// D = A(16x4,f32) * B(4x16,f32) + C(16x16,f32)   -> v_wmma_f32_16x16x4_f32
__device__ __forceinline__ v8f wmma4(v2f a, v2f b, v8f c) {
  return __builtin_amdgcn_wmma_f32_16x16x4_f32(
      /*neg_a=*/false, a, /*neg_b=*/false, b,
      /*c_mod=*/(short)0, c, /*reuse_a=*/false, /*reuse_b=*/false);
}

// ============================== KNN =========================================
__global__ __launch_bounds__(256) void knn_kernel(
    const float* __restrict__ pf,    // pos_flipped (B,N,3)
    const float* __restrict__ seed,  // (B,M,3)
    int* __restrict__ idx_out)       // (B,N,K)
{
  __shared__ float ss[M_ * 3];
  int b = blockIdx.x / (N_ / 256);
  int i = (blockIdx.x % (N_ / 256)) * 256 + threadIdx.x;
  for (int t = threadIdx.x; t < M_ * 3; t += 256)
    ss[t] = seed[(size_t)b * M_ * 3 + t];
  __syncthreads();

  float qx = pf[((size_t)b * N_ + i) * 3 + 0];
  float qy = pf[((size_t)b * N_ + i) * 3 + 1];
  float qz = pf[((size_t)b * N_ + i) * 3 + 2];

  float bd[K_];
  int bi[K_];
#pragma unroll
  for (int t = 0; t < K_; ++t) { bd[t] = 3.4e38f; bi[t] = 0; }

  for (int j = 0; j < M_; ++j) {
    float dx = qx - ss[j * 3 + 0];
    float dy = qy - ss[j * 3 + 1];
    float dz = qz - ss[j * 3 + 2];
    float d = dx * dx + dy * dy + dz * dz;
    if (d < bd[K_ - 1]) {
      bd[K_ - 1] = d; bi[K_ - 1] = j;
#pragma unroll
      for (int t = K_ - 1; t > 0; --t) {
        if (bd[t] < bd[t - 1]) {
          float td = bd[t]; bd[t] = bd[t - 1]; bd[t - 1] = td;
          int ti = bi[t];  bi[t] = bi[t - 1];  bi[t - 1] = ti;
        }
      }
    }
  }
#pragma unroll
  for (int t = 0; t < K_; ++t)
    idx_out[((size_t)b * N_ + i) * K_ + t] = bi[t];
}

// ======================= generic WMMA f32 GEMM ==============================
// Y[b,o,p] = sum_c W[o,c]*X[b,c,p] + bias[o] (+ residual[b,o,p])
__global__ __launch_bounds__(256) void gemm_bias_kernel(
    const float* __restrict__ X, const float* __restrict__ W,
    const float* __restrict__ bias, const float* __restrict__ residual,
    float* __restrict__ Y, int Cin, int Cout, int P, int totalTiles)
{
  int wave = threadIdx.x >> 5;
  int l = threadIdx.x & 31;
  int lh = l & 15, hi = l >> 4;
  int tile = blockIdx.x * 8 + wave;
  if (tile >= totalTiles) return;

  int ptiles = P >> 4;
  int otiles = Cout >> 4;
  int b  = tile / (otiles * ptiles);
  int r2 = tile % (otiles * ptiles);
  int o0 = (r2 / ptiles) << 4;
  int p0 = (r2 % ptiles) << 4;

  v8f acc = zero8();
  const float* Wb = W + (size_t)(o0 + lh) * Cin + hi * 2;
#pragma unroll 4
  for (int c0 = 0; c0 < Cin; c0 += 4) {
    v2f a;  a.x = Wb[c0];  a.y = Wb[c0 + 1];
    const float* xp = X + ((size_t)b * Cin + c0 + hi * 2) * P + p0 + lh;
    v2f bb; bb.x = xp[0]; bb.y = xp[P];
    acc = wmma4(a, bb, acc);
  }
#pragma unroll
  for (int r = 0; r < 8; ++r) {
    int ch = o0 + r + hi * 8;
    size_t oi = ((size_t)b * Cout + ch) * P + p0 + lh;
    float v = acc[r] + bias[ch];
    if (residual) v += residual[oi];
    Y[oi] = v;
  }
}

// ===================== fused per-point attention ============================
// 1 wave per (batch, point). Columns of every GEMM tile = the 16 neighbors.
// LDS staged transposed [k][ch] with padded strides so every WMMA B-operand
// is one conflict-free ds_load_b64.
__global__ __launch_bounds__(32) void attn_fused_kernel(
    const float* __restrict__ pos,    // (B,3,N)
    const float* __restrict__ seed,   // (B,M,3)
    const float* __restrict__ query,  // (B,D,N)
    const float* __restrict__ key_f,  // (B,D,M)
    const float* __restrict__ val_f,  // (B,D,M)
    const int*   __restrict__ idx,    // (B,N,K)
    const float* __restrict__ pos_w1, const float* __restrict__ pos_b1,
    const float* __restrict__ pos_g1, const float* __restrict__ pos_beta1,
    const float* __restrict__ pos_mu1, const float* __restrict__ pos_var1,
    const float* __restrict__ pos_w2, const float* __restrict__ pos_b2,
    const float* __restrict__ attn_w1, const float* __restrict__ attn_b1,
    const float* __restrict__ attn_g1, const float* __restrict__ attn_beta1,
    const float* __restrict__ attn_mu1, const float* __restrict__ attn_var1,
    const float* __restrict__ attn_w2, const float* __restrict__ attn_b2,
    float* __restrict__ agg)          // (B,D,N)
{
  __shared__ float sh_hT[K_][GS_];     // h^T:   [k][dis,rx,ry,rz]
  __shared__ float sh_pe1T[K_][HS_];   // pe1^T: [k][64]
  __shared__ float sh_xT[K_][XS_];     // (qk_rel+pe)^T: [k][256]
  __shared__ float sh_vpT[K_][XS_];    // (val_g+pe)^T:  [k][256]
  __shared__ float sh_hidT[K_][HS_];   // hid chunk^T:   [k][64]
  __shared__ int   sh_idx[K_];

  int b = blockIdx.x / N_;
  int i = blockIdx.x % N_;
  int l = threadIdx.x;
  int lh = l & 15, hi = l >> 4;

  // ---- step 1: neighbor geometry -> h (4 x 16) ----
  if (hi == 0) {
    int j = idx[((size_t)b * N_ + i) * K_ + lh];
    sh_idx[lh] = j;
    float px = pos[(size_t)b * 3 * N_ + 0 * N_ + i];
    float py = pos[(size_t)b * 3 * N_ + 1 * N_ + i];
    float pz = pos[(size_t)b * 3 * N_ + 2 * N_ + i];
    const float* sp = seed + ((size_t)b * M_ + j) * 3;
    float rx = px - sp[0], ry = py - sp[1], rz = pz - sp[2];
    sh_hT[lh][0] = sqrtf(rx * rx + ry * ry + rz * rz);
    sh_hT[lh][1] = rx; sh_hT[lh][2] = ry; sh_hT[lh][3] = rz;
  }
  __syncthreads();  // single wave: S_NOP, acts as compiler fence

  // ---- step 2: pe1 = relu(bn(pos_w1 @ h + pos_b1))   (64 x 16) ----
#pragma unroll
  for (int rt = 0; rt < 4; ++rt) {
    int o0 = rt * 16;
    v2f a;  a.x = pos_w1[(o0 + lh) * 4 + hi * 2];
            a.y = pos_w1[(o0 + lh) * 4 + hi * 2 + 1];
    v2f bb = *(const v2f*)&sh_hT[lh][hi * 2];
    v8f acc = wmma4(a, bb, zero8());
#pragma unroll
    for (int r = 0; r < 8; ++r) {
      int ch = o0 + r + hi * 8;
      float inv = pos_g1[ch] * rsqrtf(pos_var1[ch] + EPS_);
      float v = (acc[r] + pos_b1[ch]) * inv + (pos_beta1[ch] - pos_mu1[ch] * inv);
      sh_pe1T[lh][ch] = fmaxf(v, 0.f);
    }
  }

  // ---- step 3: pe = pos_w2 @ pe1 + pos_b2 (256 x 16) -> seed sh_x & sh_vp ----
  for (int rt = 0; rt < 16; ++rt) {
    int o0 = rt * 16;
    v8f acc = zero8();
#pragma unroll
    for (int kk = 0; kk < 16; ++kk) {
      int c0 = kk * 4;
      const float* wp = pos_w2 + (size_t)(o0 + lh) * PH_ + c0 + hi * 2;
      v2f a;  a.x = wp[0]; a.y = wp[1];
      v2f bb = *(const v2f*)&sh_pe1T[lh][c0 + hi * 2];
      acc = wmma4(a, bb, acc);
    }
#pragma unroll
    for (int r = 0; r < 8; ++r) {
      int ch = o0 + r + hi * 8;
      float pe = acc[r] + pos_b2[ch];
      sh_xT[lh][ch]  = pe;
      sh_vpT[lh][ch] = pe;
    }
  }

  // ---- step 4: sh_x += query - key_g ; sh_vp += val_g ----
  for (int t = l; t < D_ * K_; t += 32) {
    int ch = t >> 4, k = t & 15;
    int j = sh_idx[k];
    float q = query[((size_t)b * D_ + ch) * N_ + i];
    sh_xT[k][ch]  += q - key_f[((size_t)b * D_ + ch) * M_ + j];
    sh_vpT[k][ch] += val_f[((size_t)b * D_ + ch) * M_ + j];
  }
  __syncthreads();

  // ---- step 5: attn MLP, hid in 64-row chunks; logits acc in VGPRs ----
  v8f lg[16];
#pragma unroll
  for (int rt = 0; rt < 16; ++rt) lg[rt] = zero8();

  for (int cc = 0; cc < 8; ++cc) {
    // prefetch next chunk's weight panels while this chunk computes
    if (cc + 1 < 8) {
#pragma unroll
      for (int q = 0; q < 4; ++q) {
        __builtin_prefetch(
            attn_w1 + ((size_t)(cc + 1) * 64 + q * 16 + lh) * D_ + hi * 128,
            0, 1);
        __builtin_prefetch(
            attn_w2 + (size_t)(q * 4 + hi * 8 + (lh >> 1)) * AH_ +
                (cc + 1) * 64 + (lh & 1) * 32,
            0, 1);
      }
    }
    // attn1 partial: hid rows [cc*64, cc*64+64)
#pragma unroll
    for (int rt = 0; rt < 4; ++rt) {
      int o0 = cc * 64 + rt * 16;
      v8f acc = zero8();
#pragma unroll 4
      for (int kk = 0; kk < 64; ++kk) {
        int c0 = kk * 4;
        const float* wp = attn_w1 + (size_t)(o0 + lh) * D_ + c0 + hi * 2;
        v2f a;  a.x = wp[0]; a.y = wp[1];
        v2f bb = *(const v2f*)&sh_xT[lh][c0 + hi * 2];
        acc = wmma4(a, bb, acc);
      }
#pragma unroll
      for (int r = 0; r < 8; ++r) {
        int ch = o0 + r + hi * 8;
        float inv = attn_g1[ch] * rsqrtf(attn_var1[ch] + EPS_);
        float v = (acc[r] + attn_b1[ch]) * inv +
                  (attn_beta1[ch] - attn_mu1[ch] * inv);
        sh_hidT[lh][rt * 16 + r + hi * 8] = fmaxf(v, 0.f);
      }
    }
    // attn2 partial: logits += attn_w2[:, cc*64 : cc*64+64] @ hid_chunk
    for (int rt = 0; rt < 16; ++rt) {
      int o0 = rt * 16;
#pragma unroll
      for (int kk = 0; kk < 16; ++kk) {
        const float* wp =
            attn_w2 + (size_t)(o0 + lh) * AH_ + cc * 64 + kk * 4 + hi * 2;
        v2f a;  a.x = wp[0]; a.y = wp[1];
        v2f bb = *(const v2f*)&sh_hidT[lh][kk * 4 + hi * 2];
        lg[rt] = wmma4(a, bb, lg[rt]);
      }
    }
  }

  // ---- step 6: softmax over K (16 lanes of a half-wave) + aggregation ----
#pragma unroll
  for (int rt = 0; rt < 16; ++rt) {
#pragma unroll
    for (int r = 0; r < 8; ++r) {
      int ch = rt * 16 + r + hi * 8;
      float v = lg[rt][r] + attn_b2[ch];
      float mx = v;
      mx = fmaxf(mx, __shfl_xor(mx, 1, 32));
      mx = fmaxf(mx, __shfl_xor(mx, 2, 32));
      mx = fmaxf(mx, __shfl_xor(mx, 4, 32));
      mx = fmaxf(mx, __shfl_xor(mx, 8, 32));
      float e = __expf(v - mx);
      float s = e;
      s += __shfl_xor(s, 1, 32);
      s += __shfl_xor(s, 2, 32);
      s += __shfl_xor(s, 4, 32);
      s += __shfl_xor(s, 8, 32);
      float w = e / s;
      float p = w * sh_vpT[lh][ch];
      p += __shfl_xor(p, 1, 32);
      p += __shfl_xor(p, 2, 32);
      p += __shfl_xor(p, 4, 32);
      p += __shfl_xor(p, 8, 32);
      if (lh == 0) agg[((size_t)b * D_ + ch) * N_ + i] = p;
    }
  }
}

// ================================ launcher ==================================
extern "C" void kernel_launch(void* const* d_in, const int* in_sizes, int n_in,
                              void* d_out, int out_size, void* d_ws,
                              size_t ws_size, hipStream_t stream) {
  (void)in_sizes; (void)n_in; (void)out_size; (void)ws_size;
  const float* pos         = (const float*)d_in[0];
  const float* pos_flipped = (const float*)d_in[1];
  const float* fea         = (const float*)d_in[2];
  const float* seed        = (const float*)d_in[3];
  const float* seed_fea    = (const float*)d_in[4];
  const float* w_start = (const float*)d_in[5];
  const float* b_start = (const float*)d_in[6];
  const float* w_key   = (const float*)d_in[7];
  const float* b_key   = (const float*)d_in[8];
  const float* w_value = (const float*)d_in[9];
  const float* b_value = (const float*)d_in[10];
  const float* w_query = (const float*)d_in[11];
  const float* b_query = (const float*)d_in[12];
  const float* pos_w1 = (const float*)d_in[13];
  const float* pos_b1 = (const float*)d_in[14];
  const float* pos_g1 = (const float*)d_in[15];
  const float* pos_beta1 = (const float*)d_in[16];
  const float* pos_mu1 = (const float*)d_in[17];
  const float* pos_var1 = (const float*)d_in[18];
  const float* pos_w2 = (const float*)d_in[19];
  const float* pos_b2 = (const float*)d_in[20];
  const float* attn_w1 = (const float*)d_in[21];
  const float* attn_b1 = (const float*)d_in[22];
  const float* attn_g1 = (const float*)d_in[23];
  const float* attn_beta1 = (const float*)d_in[24];
  const float* attn_mu1 = (const float*)d_in[25];
  const float* attn_var1 = (const float*)d_in[26];
  const float* attn_w2 = (const float*)d_in[27];
  const float* attn_b2 = (const float*)d_in[28];
  const float* w_end = (const float*)d_in[29];
  const float* b_end = (const float*)d_in[30];

  // workspace layout (floats): value | key | val | query | agg | idx(int)
  float* ws      = (float*)d_ws;
  float* f_value = ws;                               // B*D*M = 1,048,576
  float* f_key   = f_value + (size_t)B_ * D_ * M_;
  float* f_val   = f_key   + (size_t)B_ * D_ * M_;
  float* f_query = f_val   + (size_t)B_ * D_ * M_;   // B*D*N = 4,194,304
  float* f_agg   = f_query + (size_t)B_ * D_ * N_;
  int*   f_idx   = (int*)(f_agg + (size_t)B_ * D_ * N_);

  // 1) KNN indices
  knn_kernel<<<B_ * (N_ / 256), 256, 0, stream>>>(pos_flipped, seed, f_idx);

  // 2) value = w_start @ seed_fea + b_start   (B,D,M)
  {
    int tiles = B_ * (D_ / 16) * (M_ / 16);
    gemm_bias_kernel<<<(tiles + 7) / 8, 256, 0, stream>>>(
        seed_fea, w_start, b_start, nullptr, f_value, C_, D_, M_, tiles);
  }
  // 3) key_f / val_f = w_key/w_value @ value + b  (B,D,M)
  {
    int tiles = B_ * (D_ / 16) * (M_ / 16);
    gemm_bias_kernel<<<(tiles + 7) / 8, 256, 0, stream>>>(
        f_value, w_key, b_key, nullptr, f_key, D_, D_, M_, tiles);
    gemm_bias_kernel<<<(tiles + 7) / 8, 256, 0, stream>>>(
        f_value, w_value, b_value, nullptr, f_val, D_, D_, M_, tiles);
  }
  // 4) query = w_query @ fea + b_query  (B,D,N)
  {
    int tiles = B_ * (D_ / 16) * (N_ / 16);
    gemm_bias_kernel<<<(tiles + 7) / 8, 256, 0, stream>>>(
        fea, w_query, b_query, nullptr, f_query, C_, D_, N_, tiles);
  }
  // 5) fused gather + pe-MLP + attn-MLP + softmax + aggregation -> agg (B,D,N)
  attn_fused_kernel<<<B_ * N_, 32, 0, stream>>>(
      pos, seed, f_query, f_key, f_val, f_idx,
      pos_w1, pos_b1, pos_g1, pos_beta1, pos_mu1, pos_var1, pos_w2, pos_b2,
      attn_w1, attn_b1, attn_g1, attn_beta1, attn_mu1, attn_var1,
      attn_w2, attn_b2, f_agg);
  // 6) out = w_end @ agg + b_end + fea  (B,C,N)
  {
    int tiles = B_ * (C_ / 16) * (N_ / 16);
    gemm_bias_kernel<<<(tiles + 7) / 8, 256, 0, stream>>>(
        f_agg, w_end, b_end, fea, (float*)d_out, D_, C_, N_, tiles);
  }
}